// QAPNet_57939108823539
// MI455X (gfx1250) — compile-verified
//
#include <hip/hip_runtime.h>
#include <math.h>

#define NN  1024
#define EE  32768
#define DD  64
#define HH  64
#define LHH 128

typedef float v2f __attribute__((ext_vector_type(2)));
typedef float v8f __attribute__((ext_vector_type(8)));

// ---------------------------------------------------------------------------
// K0: tiny MLP collapse.  v1[d], base[h], delta[h].
//   v0 = relu(be1)@We2+be2 ; v1 = relu(We1+be1)@We2+be2
//   base = N*v0@Wn + bn    ; delta = (v1-v0)@Wn
// ---------------------------------------------------------------------------
__global__ void k_prep(const float* __restrict__ We1, const float* __restrict__ be1,
                       const float* __restrict__ We2, const float* __restrict__ be2,
                       const float* __restrict__ Wn,  const float* __restrict__ bn,
                       float* __restrict__ v1_out, float* __restrict__ base_out,
                       float* __restrict__ delta_out) {
  __shared__ float sv0[DD], sv1[DD];
  int d = threadIdx.x;  // 0..63
  float s0 = be2[d], s1 = be2[d];
  for (int k = 0; k < DD; ++k) {
    float h0 = fmaxf(be1[k], 0.f);
    float h1 = fmaxf(We1[k] + be1[k], 0.f);
    float w  = We2[k * DD + d];
    s0 = fmaf(h0, w, s0);
    s1 = fmaf(h1, w, s1);
  }
  sv0[d] = s0; sv1[d] = s1; v1_out[d] = s1;
  __syncthreads();
  float b = bn[d], dl = 0.f;
  for (int k = 0; k < DD; ++k) {
    float w = Wn[k * HH + d];
    b  = fmaf((float)NN * sv0[k], w, b);
    dl = fmaf(sv1[k] - sv0[k], w, dl);
  }
  base_out[d] = b; delta_out[d] = dl;
}

// ---------------------------------------------------------------------------
// K1: zero the two 1MB transposed adjacency byte-maps (must re-zero each call)
// ---------------------------------------------------------------------------
__global__ void k_zero(unsigned int* __restrict__ p, int n_words) {
  int i = blockIdx.x * blockDim.x + threadIdx.x;
  if (i < n_words) p[i] = 0u;
}

// ---------------------------------------------------------------------------
// K2: scatter edges into transposed adjacency: connT[dst*N + src] = 1.
// Idempotent byte stores -> duplicates collapse exactly like .at[].set(1.0).
// ---------------------------------------------------------------------------
__global__ void k_scatter(const int* __restrict__ eiA, const int* __restrict__ eiB,
                          unsigned char* __restrict__ connA,
                          unsigned char* __restrict__ connB) {
  int e = blockIdx.x * blockDim.x + threadIdx.x;
  if (e >= 2 * EE) return;
  if (e < EE) {
    int s = eiA[e], d2 = eiA[EE + e];
    connA[(size_t)d2 * NN + s] = 1;
  } else {
    int le = e - EE;
    int s = eiB[le], d2 = eiB[EE + le];
    connB[(size_t)d2 * NN + s] = 1;
  }
}

// ---------------------------------------------------------------------------
// K3: per-dst distinct-src count (coalesced row sum of transposed map),
// then x[dst] = base + c*delta.  One block (256 thr) per (graph,dst).
// ---------------------------------------------------------------------------
__global__ void k_degree_x(const unsigned char* __restrict__ connA,
                           const unsigned char* __restrict__ connB,
                           const float* __restrict__ base, const float* __restrict__ delta,
                           float* __restrict__ xa, float* __restrict__ xb) {
  __shared__ float red[256];
  int b   = blockIdx.x;
  int g   = b >> 10;        // graph select
  int dst = b & (NN - 1);
  const unsigned char* conn = g ? connB : connA;
  float* x = g ? xb : xa;
  const unsigned int* row = (const unsigned int*)(conn + (size_t)dst * NN);
  unsigned int v = row[threadIdx.x];            // 4 bytes, each 0/1
  red[threadIdx.x] = (float)((v * 0x01010101u) >> 24);
  __syncthreads();
  for (int off = 128; off > 0; off >>= 1) {
    if (threadIdx.x < off) red[threadIdx.x] += red[threadIdx.x + off];
    __syncthreads();
  }
  float c = red[0];
  if (threadIdx.x < HH)
    x[(size_t)dst * HH + threadIdx.x] = fmaf(c, delta[threadIdx.x], base[threadIdx.x]);
}

// ---------------------------------------------------------------------------
// K4: P = x_a[nodes_a] @ Wl1[:64] + bl1 ; Q = x_b[nodes_b] @ Wl1[64:]
// via V_WMMA_F32_16X16X4_F32 (exact fp32).  One wave = one 16x16 tile.
// A 16x4 layout: lanes0-15 K={0,1}, lanes16-31 K={2,3}; B mirrored on N;
// C: VGPR r -> M = r + 8*(lane>=16), N = lane%16.
// ---------------------------------------------------------------------------
__global__ void k_pq_wmma(const float* __restrict__ xa, const float* __restrict__ xb,
                          const int* __restrict__ nodes_a, const int* __restrict__ nodes_b,
                          const float* __restrict__ Wl1, const float* __restrict__ bl1,
                          float* __restrict__ P, float* __restrict__ Q) {
  int id   = blockIdx.x;       // 0..1023
  int mat  = id >> 9;          // 0 -> P, 1 -> Q
  int rem  = id & 511;
  int it   = rem >> 3;         // i-tile 0..63
  int kt   = rem & 7;          // k-tile 0..7
  int i0   = it * 16, k0 = kt * 16;
  int lane = threadIdx.x;      // 0..31
  int gsel = lane >> 4;        // 0/1
  int mn   = lane & 15;

  const float* X     = mat ? xb : xa;
  const int*   nodes = mat ? nodes_b : nodes_a;
  const float* W     = Wl1 + (mat ? (size_t)HH * LHH : 0);
  float*       OUT   = mat ? Q : P;

  int row = nodes[i0 + mn];
  const float* xr = X + (size_t)row * HH;

  v8f c = {0.f, 0.f, 0.f, 0.f, 0.f, 0.f, 0.f, 0.f};
  for (int h0 = 0; h0 < HH; h0 += 4) {
    v2f a = *(const v2f*)(xr + h0 + 2 * gsel);   // A[M=mn][K=h0+2g .. +1]
    v2f bb;
    bb.x = W[(size_t)(h0 + 2 * gsel)     * LHH + k0 + mn];  // B[K][N=mn]
    bb.y = W[(size_t)(h0 + 2 * gsel + 1) * LHH + k0 + mn];
    c = __builtin_amdgcn_wmma_f32_16x16x4_f32(false, a, false, bb,
                                              (short)0, c, false, false);
  }
  float bias = mat ? 0.f : bl1[k0 + mn];
  for (int r = 0; r < 8; ++r)
    OUT[(size_t)(i0 + r + 8 * gsel) * LHH + k0 + mn] = c[r] + bias;
}

// ---------------------------------------------------------------------------
// K5: edge_attr_a / edge_attr_b are every-row == v1 (conn==1 at every edge).
// ---------------------------------------------------------------------------
__global__ void k_edge_fill(const float* __restrict__ v1, float* __restrict__ out_edges) {
  int i = blockIdx.x * blockDim.x + threadIdx.x;   // over 2*E*D
  out_edges[i] = v1[i & (DD - 1)];
}

// ---------------------------------------------------------------------------
// K6: link_probs[i,j] = sigmoid(bl2 + sum_k relu(P[i,k]+Q[j,k]) * Wl2[k])
// Block = 16x16 output tile, P/Q rows staged in LDS, float4 inner loop.
// ---------------------------------------------------------------------------
__global__ void k_link(const float* __restrict__ P, const float* __restrict__ Q,
                       const float* __restrict__ Wl2, const float* __restrict__ bl2,
                       float* __restrict__ out) {
  __shared__ float sP[16 * LHH];
  __shared__ float sQ[16 * LHH];
  __shared__ float sW[LHH];
  int tI = blockIdx.x >> 6, tJ = blockIdx.x & 63;
  int t = threadIdx.x;
  for (int s = t; s < 16 * LHH; s += 256) {
    sP[s] = P[(size_t)tI * 16 * LHH + s];
    sQ[s] = Q[(size_t)tJ * 16 * LHH + s];
  }
  if (t < LHH) sW[t] = Wl2[t];
  __syncthreads();
  int ti = t >> 4, tj = t & 15;
  const float* pr = sP + ti * LHH;
  const float* qr = sQ + tj * LHH;
  float acc = bl2[0];
  for (int k = 0; k < LHH; k += 4) {
    float4 p = *(const float4*)(pr + k);
    float4 q = *(const float4*)(qr + k);
    float4 w = *(const float4*)(sW + k);
    acc = fmaf(fmaxf(p.x + q.x, 0.f), w.x, acc);
    acc = fmaf(fmaxf(p.y + q.y, 0.f), w.y, acc);
    acc = fmaf(fmaxf(p.z + q.z, 0.f), w.z, acc);
    acc = fmaf(fmaxf(p.w + q.w, 0.f), w.w, acc);
  }
  float o = 1.f / (1.f + __expf(-acc));
  out[((size_t)tI * 16 + ti) * NN + tJ * 16 + tj] = o;
}

// ---------------------------------------------------------------------------
extern "C" void kernel_launch(void* const* d_in, const int* in_sizes, int n_in,
                              void* d_out, int out_size, void* d_ws, size_t ws_size,
                              hipStream_t stream) {
  const int*   eiA    = (const int*)  d_in[0];
  const int*   eiB    = (const int*)  d_in[1];
  const int*   nodesA = (const int*)  d_in[2];
  const int*   nodesB = (const int*)  d_in[3];
  const float* We1    = (const float*)d_in[4];
  const float* be1    = (const float*)d_in[5];
  const float* We2    = (const float*)d_in[6];
  const float* be2    = (const float*)d_in[7];
  const float* Wn     = (const float*)d_in[8];
  const float* bn     = (const float*)d_in[9];
  const float* Wl1    = (const float*)d_in[10];
  const float* bl1    = (const float*)d_in[11];
  const float* Wl2    = (const float*)d_in[12];
  const float* bl2    = (const float*)d_in[13];
  float* out = (float*)d_out;

  // workspace layout
  float* ws    = (float*)d_ws;
  float* v1    = ws;                     //   64
  float* base  = ws + 64;                //   64
  float* delta = ws + 128;               //   64
  float* xa    = ws + 192;               // 1024*64
  float* xb    = xa + NN * HH;           // 1024*64
  float* P     = xb + NN * HH;           // 1024*128
  float* Q     = P + NN * LHH;           // 1024*128
  unsigned char* connA = (unsigned char*)(Q + NN * LHH);   // 1 MB
  unsigned char* connB = connA + (size_t)NN * NN;          // 1 MB

  k_prep<<<1, 64, 0, stream>>>(We1, be1, We2, be2, Wn, bn, v1, base, delta);
  k_zero<<<(2 * NN * NN / 4) / 256, 256, 0, stream>>>((unsigned int*)connA, 2 * NN * NN / 4);
  k_scatter<<<(2 * EE + 255) / 256, 256, 0, stream>>>(eiA, eiB, connA, connB);
  k_degree_x<<<2 * NN, 256, 0, stream>>>(connA, connB, base, delta, xa, xb);
  k_pq_wmma<<<1024, 32, 0, stream>>>(xa, xb, nodesA, nodesB, Wl1, bl1, P, Q);
  k_edge_fill<<<(2 * EE * DD) / 256, 256, 0, stream>>>(v1, out + (size_t)NN * NN);
  k_link<<<(NN / 16) * (NN / 16), 256, 0, stream>>>(P, Q, Wl2, bl2, out);
}